// GCNModelGumbel_13804024889381
// MI455X (gfx1250) — compile-verified
//
#include <hip/hip_runtime.h>
#include <hip/hip_bf16.h>
#include <math.h>

typedef __attribute__((ext_vector_type(2))) float v2f;
typedef __attribute__((ext_vector_type(8))) float v8f;

#define DDIM 128
#define KDIM 64
#define NNEG 5
#define TILE 16
#define WAVES 8
#define TPB (WAVES * 32)

__device__ __forceinline__ float log_sigmoid(float x) {
    // stable: min(x,0) - log1p(exp(-|x|))
    return fminf(x, 0.0f) - log1pf(__expf(-fabsf(x)));
}

// Softmax over the 64 columns of a 16x64 tile held in WMMA C-layout
// (lane L, vgpr j of tile t holds element (row = j + 8*(L/16), col = t*16 + L%16)).
__device__ __forceinline__ void softmax_write(const v8f* acc, float* __restrict__ dst,
                                              int i0, int lo, int hi) {
#pragma unroll
    for (int j = 0; j < 8; ++j) {
        float mx = -3.402823466e+38f;
#pragma unroll
        for (int t = 0; t < 4; ++t) mx = fmaxf(mx, acc[t][j]);
#pragma unroll
        for (int m = 1; m < 16; m <<= 1) mx = fmaxf(mx, __shfl_xor(mx, m, 32));
        float e[4];
        float s = 0.0f;
#pragma unroll
        for (int t = 0; t < 4; ++t) { e[t] = __expf(acc[t][j] - mx); s += e[t]; }
#pragma unroll
        for (int m = 1; m < 16; m <<= 1) s += __shfl_xor(s, m, 32);
        const float inv = 1.0f / s;
        const int row = i0 + j + 8 * hi;
#pragma unroll
        for (int t = 0; t < 4; ++t)
            dst[(size_t)row * KDIM + t * 16 + lo] = e[t] * inv;  // coalesced across lo
    }
}

__global__ __launch_bounds__(TPB)
void gumbel_fused_kernel(const int* __restrict__ w, const int* __restrict__ c,
                         const int* __restrict__ neg,
                         const float* __restrict__ node_emb,
                         const float* __restrict__ ctx_emb,
                         const float* __restrict__ comm_w,
                         float* __restrict__ out,        // [1 + B*K + B*K]
                         float* __restrict__ tile_loss,  // [B/TILE]
                         int B)
{
    __shared__ float lds_comm[KDIM * DDIM];          // 32 KB, shared by block
    __shared__ float lds_q[WAVES][TILE * KDIM];      // 32 KB, per-wave q tiles

    // ---- stage comm_w [64x128] into LDS once per block ----
    for (int i = threadIdx.x; i < KDIM * DDIM / 4; i += TPB)
        ((float4*)lds_comm)[i] = ((const float4*)comm_w)[i];
    __syncthreads();

    const int lane = threadIdx.x & 31;
    const int lo   = lane & 15;   // column / row-in-tile selector
    const int hi   = lane >> 4;   // K-pair selector in A/B fragments
    const int wave = threadIdx.x >> 5;
    const int tile = blockIdx.x * WAVES + wave;
    const int i0   = tile * TILE;

    // ================= Phase 1: q = (we*ce) @ W^T,  priorLogit = we @ W^T ====
    // A fragment (16x4 f32): lane holds row m = lo, K elems d0, d0+1 (d0 = 4*step + 2*hi).
    const float* wrow = node_emb + (size_t)w[i0 + lo] * DDIM;
    const float* crow = node_emb + (size_t)c[i0 + lo] * DDIM;

    v8f accQ[4], accP[4];
#pragma unroll
    for (int t = 0; t < 4; ++t) { accQ[t] = (v8f)0.0f; accP[t] = (v8f)0.0f; }

    for (int step = 0; step < DDIM / 4; ++step) {
        const int d0 = step * 4 + 2 * hi;
        v2f wa = *(const v2f*)(wrow + d0);           // global_load_b64 gather
        v2f ca = *(const v2f*)(crow + d0);
        v2f aQ = wa * ca;                            // p = we*ce fragment
#pragma unroll
        for (int t = 0; t < 4; ++t) {
            // B fragment (4x16): lane holds col n = lo, K elems d0+{0,1}
            v2f b = *(const v2f*)&lds_comm[(t * 16 + lo) * DDIM + d0];
            accQ[t] = __builtin_amdgcn_wmma_f32_16x16x4_f32(
                false, aQ, false, b, (short)0, accQ[t], false, false);
            accP[t] = __builtin_amdgcn_wmma_f32_16x16x4_f32(
                false, wa, false, b, (short)0, accP[t], false, false);
        }
    }

    // ---- spill q tile to LDS (row-major 16x64) for A-layout reload ----
    float* qrow = lds_q[wave];
#pragma unroll
    for (int t = 0; t < 4; ++t)
#pragma unroll
        for (int j = 0; j < 8; ++j)
            qrow[(j + 8 * hi) * KDIM + t * 16 + lo] = accQ[t][j];

    // ---- softmax(q) and prior = softmax(we@W^T) straight from C-layout ----
    softmax_write(accQ, out + 1, i0, lo, hi);
    softmax_write(accP, out + 1 + (size_t)B * KDIM, i0, lo, hi);

    // ================= Phase 2: r[16x128] = q[16x64] @ W[64x128] ============
    v8f accR[8];
#pragma unroll
    for (int dt = 0; dt < 8; ++dt) accR[dt] = (v8f)0.0f;

    for (int ks = 0; ks < KDIM / 4; ++ks) {
        const int k0 = ks * 4 + 2 * hi;
        v2f a = *(const v2f*)&qrow[lo * KDIM + k0];  // ds_load_b64 (in-order LDS)
#pragma unroll
        for (int dt = 0; dt < 8; ++dt) {
            v2f b;
            b.x = lds_comm[(k0)     * DDIM + dt * 16 + lo];
            b.y = lds_comm[(k0 + 1) * DDIM + dt * 16 + lo];
            accR[dt] = __builtin_amdgcn_wmma_f32_16x16x4_f32(
                false, a, false, b, (short)0, accR[dt], false, false);
        }
    }

    // ================= Phase 3: pos = r.c_ctx, negp[n] = r.neg_e[n] =========
    // r C-layout: lane holds r[row = j+8*hi][d = dt*16+lo] = accR[dt][j]
    float pos[8], ng[8][NNEG];
#pragma unroll
    for (int j = 0; j < 8; ++j) {
        const int row = i0 + j + 8 * hi;
        const float* cc = ctx_emb + (size_t)c[row] * DDIM;
        float p = 0.0f;
#pragma unroll
        for (int dt = 0; dt < 8; ++dt) p += accR[dt][j] * cc[dt * 16 + lo];
        pos[j] = p;
#pragma unroll
        for (int n = 0; n < NNEG; ++n) {
            const float* nr = ctx_emb + (size_t)neg[(size_t)row * NNEG + n] * DDIM;
            float s = 0.0f;
#pragma unroll
            for (int dt = 0; dt < 8; ++dt) s += accR[dt][j] * nr[dt * 16 + lo];
            ng[j][n] = s;
        }
    }
    // reduce over the 16 lanes (d-columns) of each half
#pragma unroll
    for (int j = 0; j < 8; ++j) {
#pragma unroll
        for (int m = 1; m < 16; m <<= 1) pos[j] += __shfl_xor(pos[j], m, 32);
#pragma unroll
        for (int n = 0; n < NNEG; ++n)
#pragma unroll
            for (int m = 1; m < 16; m <<= 1) ng[j][n] += __shfl_xor(ng[j][n], m, 32);
    }

    float half_sum = 0.0f;
#pragma unroll
    for (int j = 0; j < 8; ++j) {
        float o = log_sigmoid(pos[j]);
#pragma unroll
        for (int n = 0; n < NNEG; ++n) o += log_sigmoid(-ng[j][n]);
        half_sum += o;
    }
    const float tot = half_sum + __shfl_xor(half_sum, 16, 32);
    if (lane == 0) tile_loss[tile] = tot;
}

// Deterministic fixed-order reduction of per-tile loss partials.
__global__ __launch_bounds__(256)
void reduce_loss_kernel(const float* __restrict__ tile_loss, int ntiles,
                        float* __restrict__ out, float invB)
{
    __shared__ float s[256];
    float a = 0.0f;
    for (int i = threadIdx.x; i < ntiles; i += 256) a += tile_loss[i];
    s[threadIdx.x] = a;
    __syncthreads();
    for (int off = 128; off > 0; off >>= 1) {
        if ((int)threadIdx.x < off) s[threadIdx.x] += s[threadIdx.x + off];
        __syncthreads();
    }
    if (threadIdx.x == 0) out[0] = -s[0] * invB;
}

extern "C" void kernel_launch(void* const* d_in, const int* in_sizes, int n_in,
                              void* d_out, int out_size, void* d_ws, size_t ws_size,
                              hipStream_t stream) {
    const int*   w        = (const int*)d_in[0];
    const int*   c        = (const int*)d_in[1];
    const int*   neg      = (const int*)d_in[2];
    /* d_in[3] = temp (unused, dead code in reference) */
    const float* node_emb = (const float*)d_in[4];
    const float* ctx_emb  = (const float*)d_in[5];
    const float* comm_w   = (const float*)d_in[6];
    float*       out      = (float*)d_out;

    const int B      = in_sizes[0];
    const int ntiles = B / TILE;            // 8192
    const int blocks = ntiles / WAVES;      // 1024
    float* tile_loss = (float*)d_ws;

    gumbel_fused_kernel<<<blocks, TPB, 0, stream>>>(w, c, neg, node_emb, ctx_emb,
                                                    comm_w, out, tile_loss, B);
    reduce_loss_kernel<<<1, 256, 0, stream>>>(tile_loss, ntiles, out,
                                              1.0f / (float)B);
}